// FeatureGrid_22454089024270
// MI455X (gfx1250) — compile-verified
//
#include <hip/hip_runtime.h>
#include <cstdint>
#include <cstddef>

#define GRID_N 128
#define NCH    32
#define NSPAT  (GRID_N * GRID_N * GRID_N)   // 2,097,152 cells
#define TSPAT  128                          // spatial tile per block
#define ROWW   130                          // LDS row stride in DWORDs (128 + 1 pad per 64)

// LDS word index of (channel c, spatial s) matching TDM pad layout:
// 1 DWORD of padding inserted after every 64 DWORDs streamed.
__device__ __forceinline__ uint32_t tidx(uint32_t c, uint32_t s) {
    return c * ROWW + s + (s >> 6);
}

// ---------------------------------------------------------------------------
// Kernel 1: transpose feature (C=32, S=2M) -> FT (S, 32) so each spatial cell
// has its 32 channels contiguous (128 B = one coalesced wave32 load).
// Load phase: one TDM tensor_load_to_lds per workgroup (2-D tile, padded LDS
// rows for conflict-free transposed reads). Fallback: async global->LDS.
// ---------------------------------------------------------------------------
__global__ __launch_bounds__(256) void FeatureGrid_transpose(
    const float* __restrict__ F, float* __restrict__ FT)
{
    __shared__ float tile[NCH * ROWW + 4];
    const uint32_t S   = NSPAT;
    const uint32_t s0  = blockIdx.x * (uint32_t)TSPAT;
    const uint32_t tid = threadIdx.x;

#if __has_builtin(__builtin_amdgcn_tensor_load_to_lds)
    // ---- load phase: single TDM DMA of the 32 x 128 f32 tile ----
    if (tid < 32u) {   // one wave issues the DMA (EXEC ignored by TDM)
        typedef uint32_t u32x4 __attribute__((ext_vector_type(4)));
        typedef int      i32x8 __attribute__((ext_vector_type(8)));
        typedef int      i32x4 __attribute__((ext_vector_type(4)));

        const uint64_t gaddr = (uint64_t)(const void*)(F + s0);
        const uint32_t laddr = (uint32_t)(uintptr_t)(&tile[0]);

        u32x4 g0;
        g0[0] = 1u;                                   // count=1 (valid), load, user mode
        g0[1] = laddr;                                // lds_addr (bytes)
        g0[2] = (uint32_t)gaddr;                      // global_addr[31:0]
        g0[3] = (uint32_t)((gaddr >> 32) & 0x01FFFFFFu) | (2u << 30); // addr[56:32] | type=2

        i32x8 g1;
        g1[0] = (2 << 16)      // data_size = 4 bytes
              | (1 << 20)      // pad_enable
              | (5 << 22)      // pad_interval: every 64 DWORDs
              | (0 << 25);     // pad_amount: 1 DWORD
        g1[1] = (int)((NSPAT & 0xFFFFu) << 16);                       // tensor_dim0[15:0]
        g1[2] = (int)(((NSPAT >> 16) & 0xFFFFu) | ((NCH & 0xFFFFu) << 16)); // dim0 hi | dim1 lo
        g1[3] = (int)(((NCH >> 16) & 0xFFFFu) | ((uint32_t)TSPAT << 16));   // dim1 hi | tile_dim0
        g1[4] = (int)(NCH & 0xFFFFu);                 // tile_dim1=32, tile_dim2=0 (2-D)
        g1[5] = (int)(NSPAT & 0xFFFFFFFFu);           // tensor_dim0_stride lo
        g1[6] = 0;                                    // stride hi | dim1_stride lo (unused)
        g1[7] = 0;

        i32x4 gz4 = {0, 0, 0, 0};
        i32x8 gz8 = {0, 0, 0, 0, 0, 0, 0, 0};
        // clang-23 / therock lane: 6-arg form (g0, g1, grp2, grp3, extra, cpol)
        __builtin_amdgcn_tensor_load_to_lds(g0, g1, gz4, gz4, gz8, 0);
        __builtin_amdgcn_s_wait_tensorcnt(0);
    }
#else
    // ---- fallback: CDNA5 async copies global -> LDS (ASYNCcnt) ----
    {
        const uint32_t sl = tid & (TSPAT - 1u);
        const uint32_t cp = tid >> 7;                 // 0..1
#pragma unroll
        for (int i = 0; i < 16; ++i) {
            const uint32_t c = cp + (uint32_t)i * 2u;
            const float* gsrc = F + (size_t)c * S + s0 + sl;
            uint32_t lds_off = (uint32_t)(uintptr_t)(&tile[tidx(c, sl)]);
            asm volatile("global_load_async_to_lds_b32 %0, %1, off"
                         :: "v"(lds_off), "v"(gsrc)
                         : "memory");
        }
        asm volatile("s_wait_asynccnt 0x0" ::: "memory");
    }
#endif
    __syncthreads();

    // ---- store phase: lane = channel, 128 B contiguous per (wave, row) ----
    const uint32_t c  = tid & 31u;
    const uint32_t sp = tid >> 5;                     // 0..7
#pragma unroll
    for (int i = 0; i < 16; ++i) {
        const uint32_t s = sp + (uint32_t)i * 8u;
        FT[(size_t)(s0 + s) * NCH + c] = tile[tidx(c, s)];
    }
}

// ---------------------------------------------------------------------------
// Kernel 2: trilinear interpolation from transposed layout.
// One wave32 per point, lane = channel. 8 coalesced 128 B gathers + 8 FMAs.
// ---------------------------------------------------------------------------
__global__ __launch_bounds__(256) void FeatureGrid_interp_T(
    const float* __restrict__ X, const float* __restrict__ FT,
    float* __restrict__ out, int n)
{
    const uint32_t gtid = blockIdx.x * 256u + threadIdx.x;
    const uint32_t p    = gtid >> 5;          // wave32 -> one point per wave
    const uint32_t lane = threadIdx.x & 31u;  // channel
    if (p >= (uint32_t)n) return;

    // p is wave-uniform: pin it to an SGPR so coordinate reads become SMEM loads
    const int ps = __builtin_amdgcn_readfirstlane((int)p);
    const float px = X[3 * ps + 0];
    const float py = X[3 * ps + 1];
    const float pz = X[3 * ps + 2];

    const float LOv = -10.0f, HIv = 10.0f;
    const float Gf  = (float)GRID_N;
    const float cx = 2.0f * (px - LOv) / (HIv - LOv) - 1.0f;
    const float cy = 2.0f * (py - LOv) / (HIv - LOv) - 1.0f;
    const float cz = 2.0f * (pz - LOv) / (HIv - LOv) - 1.0f;
    const float fx = ((cx + 1.0f) * Gf - 1.0f) * 0.5f;
    const float fy = ((cy + 1.0f) * Gf - 1.0f) * 0.5f;
    const float fz = ((cz + 1.0f) * Gf - 1.0f) * 0.5f;

    const float x0f = floorf(fx), y0f = floorf(fy), z0f = floorf(fz);
    const float wx = fx - x0f, wy = fy - y0f, wz = fz - z0f;
    const int x0 = (int)x0f, y0 = (int)y0f, z0 = (int)z0f;

    float acc = 0.0f;
#pragma unroll
    for (int dz = 0; dz < 2; ++dz) {
#pragma unroll
        for (int dy = 0; dy < 2; ++dy) {
#pragma unroll
            for (int dx = 0; dx < 2; ++dx) {
                const int xi = x0 + dx, yi = y0 + dy, zi = z0 + dz;
                const bool valid = ((unsigned)xi < (unsigned)GRID_N) &
                                   ((unsigned)yi < (unsigned)GRID_N) &
                                   ((unsigned)zi < (unsigned)GRID_N);
                const int xc = min(max(xi, 0), GRID_N - 1);
                const int yc = min(max(yi, 0), GRID_N - 1);
                const int zc = min(max(zi, 0), GRID_N - 1);
                const uint32_t cell = ((uint32_t)zc * GRID_N + (uint32_t)yc) * GRID_N
                                    + (uint32_t)xc;
                const float v = FT[(size_t)cell * NCH + lane];  // coalesced 128 B
                float w = (dx ? wx : 1.0f - wx) *
                          (dy ? wy : 1.0f - wy) *
                          (dz ? wz : 1.0f - wz);
                w = valid ? w : 0.0f;
                acc = fmaf(w, v, acc);
            }
        }
    }
    // write-once output: keep it out of L2 so the feature grid stays resident
    __builtin_nontemporal_store(acc, &out[(size_t)p * NCH + lane]);
}

// ---------------------------------------------------------------------------
// Fallback (ws too small): interpolate straight from channel-major layout.
// ---------------------------------------------------------------------------
__global__ __launch_bounds__(256) void FeatureGrid_interp_direct(
    const float* __restrict__ X, const float* __restrict__ F,
    float* __restrict__ out, int n)
{
    const uint32_t gtid = blockIdx.x * 256u + threadIdx.x;
    const uint32_t p    = gtid >> 5;
    const uint32_t lane = threadIdx.x & 31u;
    if (p >= (uint32_t)n) return;

    const int ps = __builtin_amdgcn_readfirstlane((int)p);
    const float px = X[3 * ps + 0];
    const float py = X[3 * ps + 1];
    const float pz = X[3 * ps + 2];

    const float LOv = -10.0f, HIv = 10.0f;
    const float Gf  = (float)GRID_N;
    const float fx = ((2.0f * (px - LOv) / (HIv - LOv)) * Gf - 1.0f) * 0.5f;
    const float fy = ((2.0f * (py - LOv) / (HIv - LOv)) * Gf - 1.0f) * 0.5f;
    const float fz = ((2.0f * (pz - LOv) / (HIv - LOv)) * Gf - 1.0f) * 0.5f;

    const float x0f = floorf(fx), y0f = floorf(fy), z0f = floorf(fz);
    const float wx = fx - x0f, wy = fy - y0f, wz = fz - z0f;
    const int x0 = (int)x0f, y0 = (int)y0f, z0 = (int)z0f;

    float acc = 0.0f;
#pragma unroll
    for (int dz = 0; dz < 2; ++dz) {
#pragma unroll
        for (int dy = 0; dy < 2; ++dy) {
#pragma unroll
            for (int dx = 0; dx < 2; ++dx) {
                const int xi = x0 + dx, yi = y0 + dy, zi = z0 + dz;
                const bool valid = ((unsigned)xi < (unsigned)GRID_N) &
                                   ((unsigned)yi < (unsigned)GRID_N) &
                                   ((unsigned)zi < (unsigned)GRID_N);
                const int xc = min(max(xi, 0), GRID_N - 1);
                const int yc = min(max(yi, 0), GRID_N - 1);
                const int zc = min(max(zi, 0), GRID_N - 1);
                const uint32_t cell = ((uint32_t)zc * GRID_N + (uint32_t)yc) * GRID_N
                                    + (uint32_t)xc;
                const float v = F[(size_t)lane * NSPAT + cell];
                float w = (dx ? wx : 1.0f - wx) *
                          (dy ? wy : 1.0f - wy) *
                          (dz ? wz : 1.0f - wz);
                w = valid ? w : 0.0f;
                acc = fmaf(w, v, acc);
            }
        }
    }
    __builtin_nontemporal_store(acc, &out[(size_t)p * NCH + lane]);
}

extern "C" void kernel_launch(void* const* d_in, const int* in_sizes, int n_in,
                              void* d_out, int out_size, void* d_ws, size_t ws_size,
                              hipStream_t stream)
{
    const float* x       = (const float*)d_in[0];   // (N, 3)
    const float* feature = (const float*)d_in[1];   // (32, 128,128,128)
    float*       out     = (float*)d_out;           // (N, 32)

    const int n = in_sizes[0] / 3;
    const size_t need = (size_t)NSPAT * NCH * sizeof(float);   // 256 MB

    const int interp_blocks = (n * 32 + 255) / 256;            // one wave per point

    if (ws_size >= need) {
        float* FT = (float*)d_ws;
        FeatureGrid_transpose<<<NSPAT / TSPAT, 256, 0, stream>>>(feature, FT);
        FeatureGrid_interp_T<<<interp_blocks, 256, 0, stream>>>(x, FT, out, n);
    } else {
        FeatureGrid_interp_direct<<<interp_blocks, 256, 0, stream>>>(x, feature, out, n);
    }
}